// SimpleGraphConv_46420006535626
// MI455X (gfx1250) — compile-verified
//
#include <hip/hip_runtime.h>
#include <hip/hip_bf16.h>
#include <stdint.h>

// ---------------------------------------------------------------------------
// SimpleGraphConv on MI455X (gfx1250, wave32)
//   h   = x @ W^T + b                      (WMMA f32 16x16x4, W staged via
//                                           global_load_async_to_lds_b128)
//   msg = segment_sum(h[src], dst)         (per-edge wave scatter, f32 atomics, L2-resident)
//   t   = h + msg / max(deg,1)             (fused with BN column stats)
//   out = relu((t-mean)*rsqrt(var+eps)*gamma + beta)
// ---------------------------------------------------------------------------

#define D_FEAT 128
#define BN_EPS 1e-5f
#define W_LDS_STRIDE 132  // 128 + 4 floats pad -> conflict-free b64 reads

typedef __attribute__((ext_vector_type(2))) float v2f;
typedef __attribute__((ext_vector_type(8))) float v8f;

// ---------------------------------------------------------------------------
// Zero workspace region (msgs, deg, column stats)
// ---------------------------------------------------------------------------
__global__ void sgc_zero_kernel(float* __restrict__ p, long long n) {
  long long i = (long long)blockIdx.x * blockDim.x + threadIdx.x;
  long long stride = (long long)gridDim.x * blockDim.x;
  for (; i < n; i += stride) p[i] = 0.0f;
}

// ---------------------------------------------------------------------------
// h = x @ W^T + b via v_wmma_f32_16x16x4_f32.
// Block = 256 threads = 8 waves. Wave w owns column tile [16w, 16w+16).
// Block b owns row tile [16b, 16b+16). 32 k-steps of K=4.
//
// W (64 KB) is staged once per block into LDS with the gfx1250 async DMA
// path (global_load_async_to_lds_b128, ASYNCcnt), padded to 132-float row
// stride: lane n in [0,16) reads dwords col*132+k0+{0,1}, lanes [16,32)
// read +{2,3} -> banks {4n,4n+1,4n+2,4n+3} mod 64, 64 distinct banks,
// zero conflicts, 8B-aligned b64 accesses.
//
// A frag (16x4): lane holds x[row0 + (lane&15)][k0 + 2*(lane>>4) + {0,1}]
// B frag (4x16): lane holds W[col0 + (lane&15)][k0 + 2*(lane>>4) + {0,1}]
// D (16x16):     vgpr v <-> row v + 8*(lane>>4), col lane&15
// ---------------------------------------------------------------------------
__global__ __launch_bounds__(256) void sgc_gemm_wmma_kernel(
    const float* __restrict__ x, const float* __restrict__ W,
    const float* __restrict__ bias, float* __restrict__ h) {
  __shared__ float wlds[D_FEAT * W_LDS_STRIDE];

  const int wave  = threadIdx.x >> 5;       // 0..7 : column tile
  const int lane  = threadIdx.x & 31;
  const int row0  = blockIdx.x << 4;        // 16 rows per block
  const int m     = lane & 15;
  const int khalf = lane >> 4;              // 0 or 1
  const int col   = (wave << 4) + m;

  // ---- async-stage W: 128 rows x 512 B = 4096 x b128 chunks, 16 per thread
  {
    const unsigned lds_base =
        (unsigned)(uintptr_t)&wlds[0];  // LDS offset lives in addr[31:0]
    const unsigned long long gw = (unsigned long long)(uintptr_t)W;
#pragma unroll
    for (int i = 0; i < 16; ++i) {
      const int c   = (int)threadIdx.x + (i << 8);
      const int row = c >> 5;            // W row
      const int off = (c & 31) << 4;     // byte offset within row [0,512)
      const unsigned           lds_addr = lds_base + row * (W_LDS_STRIDE * 4) + off;
      const unsigned long long gaddr    = gw + (unsigned long long)(row * 512 + off);
      asm volatile("global_load_async_to_lds_b128 %0, %1, off"
                   :: "v"(lds_addr), "v"(gaddr)
                   : "memory");
    }
    asm volatile("s_wait_asynccnt 0x0" ::: "memory");
  }
  __syncthreads();

  const float* __restrict__ xrow = x + (long long)(row0 + m) * D_FEAT;
  const float* __restrict__ wl   = &wlds[col * W_LDS_STRIDE];

  v8f c = {};
#pragma unroll
  for (int k0 = 0; k0 < D_FEAT; k0 += 4) {
    v2f a  = *(const v2f*)(xrow + k0 + (khalf << 1));
    v2f bf = *(const v2f*)(wl + k0 + (khalf << 1));   // ds_load_b64, conflict-free
    // D = A x B + C  (fp32 WMMA, round-to-nearest-even)
    c = __builtin_amdgcn_wmma_f32_16x16x4_f32(
        /*neg_a=*/false, a, /*neg_b=*/false, bf,
        /*c_mod=*/(short)0, c, /*reuse_a=*/false, /*reuse_b=*/false);
  }

  const float bb    = bias[col];
  const int   mbase = row0 + (khalf << 3);
#pragma unroll
  for (int v = 0; v < 8; ++v) {
    h[(long long)(mbase + v) * D_FEAT + col] = c[v] + bb;
  }
}

// ---------------------------------------------------------------------------
// Per-edge scatter: one wave32 per edge. Lane l owns features [4l, 4l+4):
// float4 gather of h[src], 4x global_atomic_add_f32 into msgs[dst].
// Lane 0 bumps deg[dst]. h/msgs (20.5 MB each) stay L2-resident (192 MB L2).
// ---------------------------------------------------------------------------
__global__ __launch_bounds__(256) void sgc_scatter_kernel(
    const int* __restrict__ ei, const float* __restrict__ h,
    float* __restrict__ msgs, float* __restrict__ deg, int nedges) {
  const int edge = (blockIdx.x << 3) + (threadIdx.x >> 5);
  const int lane = threadIdx.x & 31;
  if (edge >= nedges) return;

  const int src = ei[edge];           // edge_index row 0
  const int dst = ei[nedges + edge];  // edge_index row 1

  const float4 hv = *(const float4*)(h + (long long)src * D_FEAT + (lane << 2));
  float* mp = msgs + (long long)dst * D_FEAT + (lane << 2);
  atomicAdd(mp + 0, hv.x);
  atomicAdd(mp + 1, hv.y);
  atomicAdd(mp + 2, hv.z);
  atomicAdd(mp + 3, hv.w);
  if (lane == 0) atomicAdd(&deg[dst], 1.0f);
}

// ---------------------------------------------------------------------------
// t = h + msgs/max(deg,1); write t to d_out (scratch); accumulate per-feature
// sum / sum-of-squares with one global atomic per (block, feature).
// blockDim = 128 -> thread f owns feature f; node loop is grid-strided.
// ---------------------------------------------------------------------------
__global__ __launch_bounds__(128) void sgc_fuse_stats_kernel(
    const float* __restrict__ h, const float* __restrict__ msgs,
    const float* __restrict__ deg, float* __restrict__ t_out,
    float* __restrict__ colsum, float* __restrict__ colsumsq, int nnodes) {
  const int f = threadIdx.x;  // 0..127
  float s = 0.0f, sq = 0.0f;
  for (int n = blockIdx.x; n < nnodes; n += gridDim.x) {
    float d = deg[n];
    d = d < 1.0f ? 1.0f : d;
    const long long idx = (long long)n * D_FEAT + f;
    const float t = h[idx] + msgs[idx] / d;
    t_out[idx] = t;
    s  += t;
    sq += t * t;
  }
  atomicAdd(&colsum[f], s);
  atomicAdd(&colsumsq[f], sq);
}

// ---------------------------------------------------------------------------
// BatchNorm finalize + ReLU, in place over d_out.
// ---------------------------------------------------------------------------
__global__ __launch_bounds__(256) void sgc_finalize_kernel(
    float* __restrict__ out, const float* __restrict__ colsum,
    const float* __restrict__ colsumsq, const float* __restrict__ gamma,
    const float* __restrict__ beta, long long total, float inv_n) {
  const long long i = (long long)blockIdx.x * blockDim.x + threadIdx.x;
  if (i >= total) return;
  const int f = (int)(i & (D_FEAT - 1));
  const float mean = colsum[f] * inv_n;
  float var = colsumsq[f] * inv_n - mean * mean;
  var = var < 0.0f ? 0.0f : var;
  const float t = out[i];
  const float o = (t - mean) * rsqrtf(var + BN_EPS) * gamma[f] + beta[f];
  out[i] = o > 0.0f ? o : 0.0f;
}

// ---------------------------------------------------------------------------
// Host-side launcher. Input order (setup_inputs dict): x, edge_index, W, b,
// gamma, beta. Workspace layout (floats):
//   [h: N*D][msgs: N*D][deg: N][colsum: 128][colsumsq: 128]
// ---------------------------------------------------------------------------
extern "C" void kernel_launch(void* const* d_in, const int* in_sizes, int n_in,
                              void* d_out, int out_size, void* d_ws, size_t ws_size,
                              hipStream_t stream) {
  const float* x      = (const float*)d_in[0];
  const int*   ei     = (const int*)d_in[1];
  const float* W      = (const float*)d_in[2];
  const float* bias   = (const float*)d_in[3];
  const float* gamma  = (const float*)d_in[4];
  const float* beta   = (const float*)d_in[5];
  float*       out    = (float*)d_out;

  const int nnodes = in_sizes[0] / D_FEAT;  // 40000
  const int nedges = in_sizes[1] / 2;       // 640000

  const long long nd = (long long)nnodes * D_FEAT;

  float* h        = (float*)d_ws;
  float* msgs     = h + nd;
  float* deg      = msgs + nd;
  float* colsum   = deg + nnodes;
  float* colsumsq = colsum + D_FEAT;

  // 1) zero msgs + deg + stats (contiguous tail of the region we use)
  const long long zcount = nd + nnodes + 2 * D_FEAT;
  sgc_zero_kernel<<<2048, 256, 0, stream>>>(msgs, zcount);

  // 2) h = x @ W^T + b   (WMMA; 2500 blocks x 8 waves, each wave one 16x16 tile)
  sgc_gemm_wmma_kernel<<<nnodes / 16, 256, 0, stream>>>(x, W, bias, h);

  // 3) edge scatter: one wave per edge (8 edges / 256-thread block)
  sgc_scatter_kernel<<<(nedges + 7) / 8, 256, 0, stream>>>(ei, h, msgs, deg, nedges);

  // 4) combine + BN column statistics (t staged in d_out)
  sgc_fuse_stats_kernel<<<480, 128, 0, stream>>>(h, msgs, deg, out, colsum,
                                                 colsumsq, nnodes);

  // 5) normalize + affine + relu, in place
  const long long total = nd;
  sgc_finalize_kernel<<<(int)((total + 255) / 256), 256, 0, stream>>>(
      out, colsum, colsumsq, gamma, beta, total, 1.0f / (float)nnodes);
}